// MHCMixing_90494960926929
// MI455X (gfx1250) — compile-verified
//
#include <hip/hip_runtime.h>
#include <stdint.h>

#define N 8192
#define LOG2E 1.44269504088896340736f
#define LN2   0.69314718055994530942f

#define AS1 __attribute__((address_space(1)))
#define AS3 __attribute__((address_space(3)))

typedef int   v4i __attribute__((vector_size(16)));
typedef float v4f __attribute__((ext_vector_type(4)));

__device__ __forceinline__ float fexp2(float x) { return __builtin_amdgcn_exp2f(x); }
__device__ __forceinline__ float flog2(float x) { return __builtin_amdgcn_logf(x); }

// ---- CDNA5 async global->LDS copy (ASYNCcnt-tracked DMA path) ----
#if __has_builtin(__builtin_amdgcn_global_load_async_to_lds_b128)
#define ASYNC_B128(gp, lp)                                          \
  __builtin_amdgcn_global_load_async_to_lds_b128(                   \
      (AS1 v4i*)(uintptr_t)(gp), (AS3 v4i*)(AS3 void*)(lp), 0, 0)
#else
#define ASYNC_B128(gp, lp) do {                                              \
    unsigned _l = (unsigned)(uintptr_t)(AS3 void*)(lp);                      \
    asm volatile("global_load_async_to_lds_b128 %0, %1, off"                 \
                 :: "v"(_l), "v"((const void*)(gp)) : "memory");             \
  } while (0)
#endif

#if __has_builtin(__builtin_amdgcn_s_wait_asynccnt)
#define WAIT_ASYNC(n) __builtin_amdgcn_s_wait_asynccnt(n)
#else
#define WAIT_ASYNC(n) asm volatile("s_wait_asynccnt %0" :: "i"(n) : "memory")
#endif

// ------------------------------------------------------------------
// c[j] = 0
__global__ __launch_bounds__(256) void sk_zero(float* __restrict__ c) {
  c[blockIdx.x * 256 + threadIdx.x] = 0.0f;
}

// r[row] = logsumexp_j (L[row,j] - c[j]); one block per row, max-then-sum
__global__ __launch_bounds__(256) void sk_row_lse(const float* __restrict__ L,
                                                  const float* __restrict__ cvec,
                                                  float* __restrict__ rvec) {
  __shared__ float red[256];
  const int t = threadIdx.x;
  const int row = blockIdx.x;
  const v4f* Lr = (const v4f*)(L + (size_t)row * N);
  const v4f* cv = (const v4f*)cvec;

  float x0[8], x1[8], x2[8], x3[8];
  float m = -__builtin_inff();
#pragma unroll
  for (int k = 0; k < 8; ++k) {
    v4f a = Lr[t + k * 256];
    v4f b = cv[t + k * 256];
    x0[k] = a.x - b.x; x1[k] = a.y - b.y; x2[k] = a.z - b.z; x3[k] = a.w - b.w;
    m = fmaxf(m, fmaxf(fmaxf(x0[k], x1[k]), fmaxf(x2[k], x3[k])));
  }
  red[t] = m; __syncthreads();
  for (int o = 128; o > 0; o >>= 1) {
    if (t < o) red[t] = fmaxf(red[t], red[t + o]);
    __syncthreads();
  }
  const float M = red[0];
  __syncthreads();

  float s = 0.0f;
#pragma unroll
  for (int k = 0; k < 8; ++k) {
    s += fexp2((x0[k] - M) * LOG2E);
    s += fexp2((x1[k] - M) * LOG2E);
    s += fexp2((x2[k] - M) * LOG2E);
    s += fexp2((x3[k] - M) * LOG2E);
  }
  red[t] = s; __syncthreads();
  for (int o = 128; o > 0; o >>= 1) {
    if (t < o) red[t] += red[t + o];
    __syncthreads();
  }
  if (t == 0) rvec[row] = M + flog2(red[0]) * LN2;
}

// Partial column LSE over a chunk of rows: online (max,sum) per column.
// Block = 256 threads x 4 columns = 1024-column strip (blockIdx.y of 8).
// Rows streamed 8 at a time through LDS via async global->LDS b128 copies,
// double-buffered on ASYNCcnt.
__global__ __launch_bounds__(256) void sk_col_partial(const float* __restrict__ L,
                                                      const float* __restrict__ rvec,
                                                      float2* __restrict__ part,
                                                      int steps) {
  __shared__ v4f buf[2][8][256];  // 64 KB of the 320 KB WGP LDS
  const int t = threadIdx.x;
  const int chunk = blockIdx.x;
  const int colBase = blockIdx.y * 1024 + t * 4;
  const int i0 = chunk * steps * 8;

  // prime step 0
#pragma unroll
  for (int rr = 0; rr < 8; ++rr)
    ASYNC_B128(L + (size_t)(i0 + rr) * N + colBase, &buf[0][rr][t]);

  float m0 = -__builtin_inff(), m1 = m0, m2 = m0, m3 = m0;
  float s0 = 0.f, s1 = 0.f, s2 = 0.f, s3 = 0.f;

  for (int st = 0; st < steps; ++st) {
    const int sb = st & 1;
    if (st + 1 < steps) {
      const int i1 = i0 + (st + 1) * 8;
#pragma unroll
      for (int rr = 0; rr < 8; ++rr)
        ASYNC_B128(L + (size_t)(i1 + rr) * N + colBase, &buf[sb ^ 1][rr][t]);
      WAIT_ASYNC(8);   // in-order: the 8 copies of step `st` are complete
    } else {
      WAIT_ASYNC(0);
    }

    const int ib = i0 + st * 8;
    float rb[8];
#pragma unroll
    for (int rr = 0; rr < 8; ++rr) rb[rr] = rvec[ib + rr];

    float x0[8], x1[8], x2[8], x3[8];
#pragma unroll
    for (int rr = 0; rr < 8; ++rr) {
      v4f w = buf[sb][rr][t];
      x0[rr] = w.x - rb[rr]; x1[rr] = w.y - rb[rr];
      x2[rr] = w.z - rb[rr]; x3[rr] = w.w - rb[rr];
    }
    // block-of-8 max per column, then one rescale of the running sums
    float c0 = x0[0], c1 = x1[0], c2 = x2[0], c3 = x3[0];
#pragma unroll
    for (int rr = 1; rr < 8; ++rr) {
      c0 = fmaxf(c0, x0[rr]); c1 = fmaxf(c1, x1[rr]);
      c2 = fmaxf(c2, x2[rr]); c3 = fmaxf(c3, x3[rr]);
    }
    float n0 = fmaxf(m0, c0), n1 = fmaxf(m1, c1), n2 = fmaxf(m2, c2), n3 = fmaxf(m3, c3);
    s0 *= fexp2((m0 - n0) * LOG2E); m0 = n0;
    s1 *= fexp2((m1 - n1) * LOG2E); m1 = n1;
    s2 *= fexp2((m2 - n2) * LOG2E); m2 = n2;
    s3 *= fexp2((m3 - n3) * LOG2E); m3 = n3;
#pragma unroll
    for (int rr = 0; rr < 8; ++rr) {
      s0 += fexp2((x0[rr] - m0) * LOG2E);
      s1 += fexp2((x1[rr] - m1) * LOG2E);
      s2 += fexp2((x2[rr] - m2) * LOG2E);
      s3 += fexp2((x3[rr] - m3) * LOG2E);
    }
  }

  part[(size_t)chunk * N + colBase + 0] = make_float2(m0, s0);
  part[(size_t)chunk * N + colBase + 1] = make_float2(m1, s1);
  part[(size_t)chunk * N + colBase + 2] = make_float2(m2, s2);
  part[(size_t)chunk * N + colBase + 3] = make_float2(m3, s3);
}

// c[col] = merge of per-chunk (max,sum) partials -> LSE
__global__ __launch_bounds__(256) void sk_col_combine(const float2* __restrict__ part,
                                                      float* __restrict__ cvec,
                                                      int nchunk) {
  const int col = blockIdx.x * 256 + threadIdx.x;
  float m = -__builtin_inff(), s = 0.f;
  for (int ch = 0; ch < nchunk; ++ch) {
    float2 p = part[(size_t)ch * N + col];
    float nm = fmaxf(m, p.x);
    s = s * fexp2((m - nm) * LOG2E) + p.y * fexp2((p.x - nm) * LOG2E);
    m = nm;
  }
  cvec[col] = m + flog2(s) * LN2;
}

// out = exp(L - r[i] - c[j]); nontemporal b128 stores (write-once stream)
__global__ __launch_bounds__(256) void sk_final(const float* __restrict__ L,
                                                const float* __restrict__ rvec,
                                                const float* __restrict__ cvec,
                                                float* __restrict__ out) {
  const size_t idx4 = (size_t)blockIdx.x * 256 + threadIdx.x;
  const int i = (int)(idx4 >> 11);      // 2048 float4 per row
  const int j4 = (int)(idx4 & 2047);
  v4f a = ((const v4f*)L)[idx4];
  v4f cc = ((const v4f*)cvec)[j4];
  float rb = rvec[i];
  v4f o;
  o.x = fexp2((a.x - rb - cc.x) * LOG2E);
  o.y = fexp2((a.y - rb - cc.y) * LOG2E);
  o.z = fexp2((a.z - rb - cc.z) * LOG2E);
  o.w = fexp2((a.w - rb - cc.w) * LOG2E);
  __builtin_nontemporal_store(o, ((v4f*)out) + idx4);
}

// ------------------------------------------------------------------
extern "C" void kernel_launch(void* const* d_in, const int* in_sizes, int n_in,
                              void* d_out, int out_size, void* d_ws, size_t ws_size,
                              hipStream_t stream) {
  (void)in_sizes; (void)n_in; (void)out_size;
  const float* L = (const float*)d_in[0];
  float* out = (float*)d_out;

  float* cvec = (float*)d_ws;            // 32 KB
  float* rvec = cvec + N;                // 32 KB
  float2* part = (float2*)((char*)d_ws + 64 * 1024);

  int nchunk = 64;                       // 4 MB of partials at 64 chunks
  while (nchunk > 1 &&
         (size_t)64 * 1024 + (size_t)nchunk * N * sizeof(float2) > ws_size)
    nchunk >>= 1;
  const int steps = (N / nchunk) / 8;    // 8 rows per pipelined step

  sk_zero<<<N / 256, 256, 0, stream>>>(cvec);
  for (int it = 0; it < 10; ++it) {
    sk_row_lse<<<N, 256, 0, stream>>>(L, cvec, rvec);
    sk_col_partial<<<dim3(nchunk, N / 1024), 256, 0, stream>>>(L, rvec, part, steps);
    sk_col_combine<<<N / 256, 256, 0, stream>>>(part, cvec, nchunk);
  }
  sk_final<<<(N * (N / 4)) / 256, 256, 0, stream>>>(L, rvec, cvec, out);
}